// EdgeClassifier_12103217840679
// MI455X (gfx1250) — compile-verified
//
#include <hip/hip_runtime.h>

// Problem constants from the reference
#define NN    100000
#define EE    1600000
#define INC   128
#define OUTC  64
#define ROW_TILES (NN / 16)          // 6250 exact
#define TOT_EDGES (2 * EE)           // pos + neg edges for classifier

typedef __attribute__((ext_vector_type(2))) float v2f;
typedef __attribute__((ext_vector_type(8))) float v8f;

// ---------------------------------------------------------------------------
// Kernel 0: init. agg[N*64] = 0, deg[n] = 1.0 (self-loop contribution).
// ---------------------------------------------------------------------------
__global__ void init_kernel(float* __restrict__ deg, float* __restrict__ agg) {
  const int tid = blockIdx.x * blockDim.x + threadIdx.x;
  if (tid < NN * OUTC) agg[tid] = 0.0f;
  if (tid < NN)        deg[tid] = 1.0f;   // self loop
}

// ---------------------------------------------------------------------------
// Kernel 1: degree accumulation over edge destinations.
// ---------------------------------------------------------------------------
__global__ void degree_kernel(const int* __restrict__ ei, float* __restrict__ deg) {
  const int e = blockIdx.x * blockDim.x + threadIdx.x;
  if (e >= EE) return;
  const int d = ei[EE + e];               // edge_index[1][e]
  atomicAdd(&deg[d], 1.0f);
}

// ---------------------------------------------------------------------------
// Kernel 2: dinv = rsqrt(max(deg, 1e-12)), in place.
// ---------------------------------------------------------------------------
__global__ void dinv_kernel(float* __restrict__ deg) {
  const int n = blockIdx.x * blockDim.x + threadIdx.x;
  if (n >= NN) return;
  deg[n] = rsqrtf(fmaxf(deg[n], 1e-12f));
}

// ---------------------------------------------------------------------------
// Kernel 3: xw = x @ W via V_WMMA_F32_16X16X4_F32 (fp32 tensor path).
// One wave handles one 16-row tile across all 64 output columns
// (4 accumulators of 16x16). K loop: 128/4 = 32 iters -> 128 WMMAs/wave.
//
// fp32 A 16x4 fragment (2 VGPRs):  lanes 0-15: M=lane,   K = k0,  k0+1
//                                  lanes16-31: M=lane-16, K = k0+2, k0+3
// fp32 B 4x16 fragment (2 VGPRs):  mirrored (row striped across lanes)
// fp32 C/D 16x16 (8 VGPRs): VGPR v -> M = v + 8*(lane>=16), N = lane&15 (+16*nt)
// ---------------------------------------------------------------------------
__global__ void xw_wmma_kernel(const float* __restrict__ x,
                               const float* __restrict__ W,
                               float* __restrict__ xw) {
  const int wave = (blockIdx.x * blockDim.x + threadIdx.x) >> 5;
  if (wave >= ROW_TILES) return;        // whole-wave uniform: EXEC stays all-1s
  const int lane = threadIdx.x & 31;
  const int half = lane >> 4;
  const int m    = lane & 15;
  const int row0 = wave * 16;

  v8f acc0 = {}, acc1 = {}, acc2 = {}, acc3 = {};
  const float* arow = x + (size_t)(row0 + m) * INC;

  for (int k0 = 0; k0 < INC; k0 += 4) {
    const int kb = k0 + 2 * half;
    v2f a;  a.x  = arow[kb];            a.y  = arow[kb + 1];
    const float* w0 = W + (size_t)kb * OUTC;
    const float* w1 = w0 + OUTC;
    v2f b0; b0.x = w0[m];               b0.y = w1[m];
    v2f b1; b1.x = w0[16 + m];          b1.y = w1[16 + m];
    v2f b2; b2.x = w0[32 + m];          b2.y = w1[32 + m];
    v2f b3; b3.x = w0[48 + m];          b3.y = w1[48 + m];
    acc0 = __builtin_amdgcn_wmma_f32_16x16x4_f32(false, a, false, b0, (short)0, acc0, false, false);
    acc1 = __builtin_amdgcn_wmma_f32_16x16x4_f32(false, a, false, b1, (short)0, acc1, false, false);
    acc2 = __builtin_amdgcn_wmma_f32_16x16x4_f32(false, a, false, b2, (short)0, acc2, false, false);
    acc3 = __builtin_amdgcn_wmma_f32_16x16x4_f32(false, a, false, b3, (short)0, acc3, false, false);
  }

  for (int v = 0; v < 8; ++v) {
    const size_t r = (size_t)(row0 + v + 8 * half) * OUTC;
    xw[r + m]        = acc0[v];
    xw[r + 16 + m]   = acc1[v];
    xw[r + 32 + m]   = acc2[v];
    xw[r + 48 + m]   = acc3[v];
  }
}

// ---------------------------------------------------------------------------
// Kernel 4: message pass. For each edge (and each self loop) scatter-add
// norm * xw[src] into agg[dst]. 64 consecutive threads share one edge ->
// coalesced xw row read, coalesced atomic row update (L2-resident).
// ---------------------------------------------------------------------------
__global__ void message_kernel(const int* __restrict__ ei,
                               const float* __restrict__ dinv,
                               const float* __restrict__ xw,
                               float* __restrict__ agg) {
  const long long tid = (long long)blockIdx.x * blockDim.x + threadIdx.x;
  const long long e   = tid >> 6;
  if (e >= (long long)(EE + NN)) return;
  const int ch = (int)(tid & 63);
  int s, d;
  if (e < EE) { s = ei[e]; d = ei[EE + e]; }
  else        { s = d = (int)(e - EE); }          // self loop
  const float norm = dinv[s] * dinv[d];
  atomicAdd(&agg[(size_t)d * OUTC + ch], norm * xw[(size_t)s * OUTC + ch]);
}

// ---------------------------------------------------------------------------
// Kernel 5: h = relu(agg + b), in place.
// ---------------------------------------------------------------------------
__global__ void relu_bias_kernel(float* __restrict__ agg, const float* __restrict__ b) {
  const int tid = blockIdx.x * blockDim.x + threadIdx.x;
  if (tid >= NN * OUTC) return;
  agg[tid] = fmaxf(agg[tid] + b[tid & (OUTC - 1)], 0.0f);
}

// ---------------------------------------------------------------------------
// Kernel 6: classifier. One wave32 per output edge; lanes stride channels
// (coalesced h-row gathers), shuffle reduction, lane 0 writes 2 logits.
// out[j] for j in [0,2E): s/d from edge_index if j<E else neg_edge_index.
// ---------------------------------------------------------------------------
__global__ void classify_kernel(const int* __restrict__ ei,
                                const int* __restrict__ nei,
                                const float* __restrict__ h,
                                const float* __restrict__ Wc,
                                const float* __restrict__ bc,
                                float* __restrict__ out) {
  const long long wave = ((long long)blockIdx.x * blockDim.x + threadIdx.x) >> 5;
  if (wave >= (long long)TOT_EDGES) return;
  const int lane = threadIdx.x & 31;
  int s, d;
  if (wave < EE) { s = ei[wave];         d = ei[EE + wave]; }
  else           { s = nei[wave - EE];   d = nei[EE + (wave - EE)]; }

  const float* hs = h + (size_t)s * OUTC;
  const float* hd = h + (size_t)d * OUTC;
  float o0 = 0.0f, o1 = 0.0f;
  for (int c = lane; c < OUTC; c += 32) {
    const float vs = hs[c];
    const float vd = hd[c];
    o0 += vs * Wc[c * 2 + 0]          + vd * Wc[(OUTC + c) * 2 + 0];
    o1 += vs * Wc[c * 2 + 1]          + vd * Wc[(OUTC + c) * 2 + 1];
  }
  for (int off = 16; off > 0; off >>= 1) {
    o0 += __shfl_down(o0, off, 32);
    o1 += __shfl_down(o1, off, 32);
  }
  if (lane == 0) {
    out[wave * 2 + 0] = o0 + bc[0];
    out[wave * 2 + 1] = o1 + bc[1];
  }
}

// ---------------------------------------------------------------------------
// Host launcher
// ---------------------------------------------------------------------------
extern "C" void kernel_launch(void* const* d_in, const int* in_sizes, int n_in,
                              void* d_out, int out_size, void* d_ws, size_t ws_size,
                              hipStream_t stream) {
  const float* x   = (const float*)d_in[0];
  const int*   ei  = (const int*)  d_in[1];
  const int*   nei = (const int*)  d_in[2];
  const float* W   = (const float*)d_in[3];
  const float* b   = (const float*)d_in[4];
  const float* Wc  = (const float*)d_in[5];
  const float* bc  = (const float*)d_in[6];
  float*       out = (float*)d_out;

  // Workspace layout (floats): dinv[N] | xw[N*64] | agg[N*64]
  float* ws   = (float*)d_ws;
  float* dinv = ws;
  float* xw   = ws + ((NN + 255) / 256) * 256;       // 100096 (aligned)
  float* agg  = xw + (size_t)NN * OUTC;

  const int BLK = 256;

  // 0: init deg + agg
  init_kernel<<<(NN * OUTC + BLK - 1) / BLK, BLK, 0, stream>>>(dinv, agg);
  // 1: degree atomics over dst
  degree_kernel<<<(EE + BLK - 1) / BLK, BLK, 0, stream>>>(ei, dinv);
  // 2: dinv = rsqrt(max(deg,eps))
  dinv_kernel<<<(NN + BLK - 1) / BLK, BLK, 0, stream>>>(dinv);
  // 3: xw = x @ W (WMMA fp32); 8 waves/block, one 16x64 tile per wave
  {
    const int waves_per_blk = BLK / 32;
    const int grid = (ROW_TILES + waves_per_blk - 1) / waves_per_blk;  // 782
    xw_wmma_kernel<<<grid, BLK, 0, stream>>>(x, W, xw);
  }
  // 4: edge + self-loop scatter-add
  {
    const long long threads = (long long)(EE + NN) * OUTC;             // 108.8M
    const long long grid = (threads + BLK - 1) / BLK;
    message_kernel<<<(unsigned)grid, BLK, 0, stream>>>(ei, dinv, xw, agg);
  }
  // 5: h = relu(agg + b)
  relu_bias_kernel<<<(NN * OUTC + BLK - 1) / BLK, BLK, 0, stream>>>(agg, b);
  // 6: classifier, one wave per output edge
  {
    const long long waves = (long long)TOT_EDGES;                      // 3.2M
    const long long grid = (waves * 32 + BLK - 1) / BLK;               // 400000
    classify_kernel<<<(unsigned)grid, BLK, 0, stream>>>(ei, nei, agg, Wc, bc, out);
  }
}